// GaussianDPMM_2027224563984
// MI455X (gfx1250) — compile-verified
//
#include <hip/hip_runtime.h>
#include <math.h>

// ---------------- problem constants (match reference) ----------------
#define KN   128      // clusters
#define DD   64       // feature dim
#define ACOL 128      // contraction length = 2*DD  ([x^2, x])
#define XSTR 66       // padded LDS stride for x rows (bank spread, 8B aligned)
#define WSTR 130      // padded LDS stride for W rows (bank spread, 8B aligned)
#define LOG2PI 1.8378770664093453f

typedef __attribute__((ext_vector_type(2))) float v2f;
typedef __attribute__((ext_vector_type(8))) float v8f;

// ---------------- device math helpers ----------------
__device__ __forceinline__ float digammaf(float x) {
  // recurrence to x >= 6, then asymptotic series
  float r = 0.f;
  while (x < 6.f) { r -= 1.f / x; x += 1.f; }
  float ix = 1.f / x, ix2 = ix * ix;
  return r + logf(x) - 0.5f * ix
         - ix2 * (1.f / 12.f - ix2 * (1.f / 120.f - ix2 * (1.f / 252.f)));
}

// ---------------- kernel 1: per-cluster prep (one block of 128) ----------------
// Writes: Wg[KN][128]  (cols 0..63 = n/B, cols 64..127 = -2*tau*n/B)
//         biasg[KN]    (e_log_pi + 0.5*(e_log_det - D*log2pi) - 0.5*(sum tau^2*nb + D/c))
//         scal[1] = total KL
__global__ __launch_bounds__(KN) void dpmm_prep(
    const float* __restrict__ nat_u, const float* __restrict__ nat_v,
    const float* __restrict__ nat_tau, const float* __restrict__ nat_c,
    const float* __restrict__ nat_n, const float* __restrict__ nat_B,
    float* __restrict__ Wg, float* __restrict__ biasg, float* __restrict__ scal) {
  __shared__ float sh[KN];
  __shared__ float red[KN];
  const int k = threadIdx.x;

  float u = nat_u[k] + 1.f;
  float v = nat_v[k] + 1.f;
  float c = nat_c[k];
  float n = nat_n[k] - (float)DD - 2.f;

  float a1 = 0.5f * n;
  float dg_a1 = digammaf(a1);
  float lg_a1 = lgammaf(a1);
  const float a0g = 0.5f * ((float)DD + 2.f);   // N0/2
  const float b0g = 0.5f;                        // B0/2
  const float lg_a0g = lgammaf(a0g);
  const float log_b0g = logf(b0g);

  float e_log_det = (float)DD * dg_a1;
  float sum_quad = 0.f, kl_gamma = 0.f, kl_norm = 0.f;
  for (int d = 0; d < DD; ++d) {
    float tau = nat_tau[k * DD + d] / c;
    float Bd  = nat_B[k * DD + d] - c * tau * tau;
    float nb  = n / Bd;
    float b1  = 0.5f * Bd;
    e_log_det -= logf(b1);
    sum_quad  += tau * tau * nb;
    Wg[k * ACOL + d]       = nb;
    Wg[k * ACOL + DD + d]  = -2.f * tau * nb;
    kl_gamma += (a1 - a0g) * dg_a1 - lg_a1 + lg_a0g
              + a0g * (logf(b1) - log_b0g) + a1 * (b0g - b1) / b1;
    kl_norm  += logf(c) + 1.f / c - 1.f + nb * tau * tau;   // C0=1, TAU0=0
  }
  kl_norm *= 0.5f;

  // stick-breaking E[log pi]: exclusive prefix of e_log_1m
  float dg_u = digammaf(u), dg_v = digammaf(v), dg_uv = digammaf(u + v);
  float e_stick = dg_u - dg_uv;
  float e_1m    = dg_v - dg_uv;
  sh[k] = e_1m;
  for (int off = 1; off < KN; off <<= 1) {
    __syncthreads();
    float t = (k >= off) ? sh[k - off] : 0.f;
    __syncthreads();
    sh[k] += t;
  }
  __syncthreads();
  float e_log_pi = e_stick + sh[k] - e_1m;   // exclusive scan

  biasg[k] = e_log_pi + 0.5f * (e_log_det - (float)DD * LOG2PI)
           - 0.5f * (sum_quad + (float)DD / c);

  // KL(Beta(u,v) || Beta(1,1)); prior lgamma terms are 0
  float kl_beta = lgammaf(u + v) - lgammaf(u) - lgammaf(v)
                + (u - 1.f) * dg_u + (v - 1.f) * dg_v + (2.f - u - v) * dg_uv;

  red[k] = kl_beta + kl_gamma + kl_norm;
  __syncthreads();
  for (int off = KN / 2; off > 0; off >>= 1) {
    if (k < off) red[k] += red[k + off];
    __syncthreads();
  }
  if (k == 0) scal[1] = red[0];
}

// ---------------- kernel 2: fused WMMA GEMM + logsumexp + normalize ----------------
// 8 waves/block, 16 rows/wave -> 128 rows/block. Each wave: 8 accumulators
// (16x16 f32 tiles) covering all 128 clusters; 32 k-steps of V_WMMA_F32_16X16X4_F32.
__global__ __launch_bounds__(256) void dpmm_estep(
    const float* __restrict__ x, const float* __restrict__ Wg,
    const float* __restrict__ biasg, float* __restrict__ r_out,
    float* __restrict__ llpart) {
  __shared__ float w_lds[KN * WSTR];        // 66560 B
  __shared__ float x_lds[128 * XSTR];       // 33792 B (8 waves x 16 rows)
  __shared__ float bias_lds[KN];
  __shared__ float ll_lds[16];

  const int tid  = threadIdx.x;
  const int lane = tid & 31;
  const int wave = tid >> 5;
  const int laneN = lane & 15;
  const int half  = lane >> 4;
  const long row0 = (long)blockIdx.x * 128;

  // stage W [128 x 128] into padded LDS (float2 granularity)
  for (int i = tid; i < KN * (ACOL / 2); i += 256) {
    int kk = i >> 6, d2 = (i & 63) << 1;
    *(v2f*)&w_lds[kk * WSTR + d2] = *(const v2f*)&Wg[kk * ACOL + d2];
  }
  if (tid < KN) bias_lds[tid] = biasg[tid];
  // stage 128 rows of x
  for (int i = tid; i < 128 * (DD / 2); i += 256) {
    int rr = i >> 5, d2 = (i & 31) << 1;
    *(v2f*)&x_lds[rr * XSTR + d2] = *(const v2f*)&x[(row0 + rr) * DD + d2];
  }
  __syncthreads();

  v8f acc[8];
  v8f zero = {0.f, 0.f, 0.f, 0.f, 0.f, 0.f, 0.f, 0.f};
#pragma unroll
  for (int t = 0; t < 8; ++t) acc[t] = zero;

  const float* xrow = &x_lds[(wave * 16 + laneN) * XSTR];
  const int h2 = half << 1;

#pragma unroll 4
  for (int kk = 0; kk < 32; ++kk) {
    const int acol = (kk << 2) + h2;      // A/B fragment d-offset per ISA layout
    const int xoff = acol & (DD - 1);
    v2f xv = *(const v2f*)&xrow[xoff];
    v2f a  = (kk < 16) ? (v2f)(xv * xv) : xv;   // cols 0..63 = x^2, 64..127 = x
#pragma unroll
    for (int t = 0; t < 8; ++t) {
      v2f b = *(const v2f*)&w_lds[(t * 16 + laneN) * WSTR + acol];
      acc[t] = __builtin_amdgcn_wmma_f32_16x16x4_f32(
          false, a, false, b, (short)0, acc[t], false, false);
    }
  }

  // fused epilogue: log_rho = bias - 0.5*S ; logsumexp over 128 clusters per row
  float biasv[8];
#pragma unroll
  for (int t = 0; t < 8; ++t) biasv[t] = bias_lds[t * 16 + laneN];

  float ll_local = 0.f;
#pragma unroll
  for (int j = 0; j < 8; ++j) {
    float lr[8];
    float mx = -3.402823466e38f;
#pragma unroll
    for (int t = 0; t < 8; ++t) {
      lr[t] = fmaf(-0.5f, acc[t][j], biasv[t]);
      mx = fmaxf(mx, lr[t]);
    }
#pragma unroll
    for (int m = 1; m < 16; m <<= 1) mx = fmaxf(mx, __shfl_xor(mx, m, 32));
    float s = 0.f;
#pragma unroll
    for (int t = 0; t < 8; ++t) { lr[t] = __expf(lr[t] - mx); s += lr[t]; }
#pragma unroll
    for (int m = 1; m < 16; m <<= 1) s += __shfl_xor(s, m, 32);
    float ln   = mx + __logf(s);
    float invs = 1.f / s;

    const long n = row0 + wave * 16 + j + half * 8;   // C/D row layout per ISA
    float* orow = &r_out[n * (long)KN];
#pragma unroll
    for (int t = 0; t < 8; ++t) orow[t * 16 + laneN] = lr[t] * invs;
    if (laneN == 0) ll_local += ln;
  }
  if (laneN == 0) ll_lds[wave * 2 + half] = ll_local;
  __syncthreads();
  if (tid == 0) {
    float s = 0.f;
#pragma unroll
    for (int i = 0; i < 16; ++i) s += ll_lds[i];
    llpart[blockIdx.x] = s;    // deterministic: no fp atomics
  }
}

// ---------------- kernel 3: deterministic reduce + finalize ----------------
__global__ __launch_bounds__(256) void dpmm_finalize(
    const float* __restrict__ llpart, int nblocks,
    const float* __restrict__ scal, float* __restrict__ out_neg_elbo) {
  __shared__ float sm[256];
  float s = 0.f;
  for (int i = threadIdx.x; i < nblocks; i += 256) s += llpart[i];
  sm[threadIdx.x] = s;
  __syncthreads();
  for (int off = 128; off > 0; off >>= 1) {
    if ((int)threadIdx.x < off) sm[threadIdx.x] += sm[threadIdx.x + off];
    __syncthreads();
  }
  if (threadIdx.x == 0) out_neg_elbo[0] = scal[1] - sm[0];  // -(ll - KL)
}

// ---------------- host launcher ----------------
extern "C" void kernel_launch(void* const* d_in, const int* in_sizes, int n_in,
                              void* d_out, int out_size, void* d_ws, size_t ws_size,
                              hipStream_t stream) {
  const float* x       = (const float*)d_in[0];
  const float* nat_u   = (const float*)d_in[1];
  const float* nat_v   = (const float*)d_in[2];
  const float* nat_tau = (const float*)d_in[3];
  const float* nat_c   = (const float*)d_in[4];
  const float* nat_n   = (const float*)d_in[5];
  const float* nat_B   = (const float*)d_in[6];
  (void)n_in; (void)ws_size;

  const int N = in_sizes[0] / DD;          // 262144
  const int nblocks = N / 128;             // 2048

  float* out = (float*)d_out;              // [N*KN] r, then [1] -elbo
  float* ws  = (float*)d_ws;
  float* scal   = ws;                      // scal[1] = KL total
  float* Wg     = ws + 16;                 // [KN*ACOL]
  float* biasg  = Wg + KN * ACOL;          // [KN]
  float* llpart = biasg + KN;              // [nblocks]

  dpmm_prep<<<1, KN, 0, stream>>>(nat_u, nat_v, nat_tau, nat_c, nat_n, nat_B,
                                  Wg, biasg, scal);
  dpmm_estep<<<nblocks, 256, 0, stream>>>(x, Wg, biasg, out, llpart);
  dpmm_finalize<<<1, 256, 0, stream>>>(llpart, nblocks, scal,
                                       out + (size_t)N * KN);
  (void)out_size;
}